// VectorQuantizer_14181982012064
// MI455X (gfx1250) — compile-verified
//
#include <hip/hip_runtime.h>
#include <hip/hip_bf16.h>

#define N_TOT   32768
#define K_CB    8192
#define D_DIM   256
#define LDSX_ST (D_DIM + 4)   // padded LDS row stride (floats) -> conflict-free A reads

typedef __attribute__((ext_vector_type(2))) float v2f;
typedef __attribute__((ext_vector_type(8))) float v8f;

// ---------------------------------------------------------------------------
// Kernel 1: per-codeword squared norms ||e_k||^2 -> ws[0..K); zero loss slot.
// One wave32 per codeword; float4 loads, shfl butterfly reduce.
// ---------------------------------------------------------------------------
__global__ __launch_bounds__(256) void vq_norms(const float* __restrict__ emb,
                                                float* __restrict__ ws) {
    if (blockIdx.x == 0 && threadIdx.x == 0) ws[K_CB] = 0.0f;
    const int wave = (blockIdx.x * blockDim.x + threadIdx.x) >> 5;
    const int lane = threadIdx.x & 31;
    if (wave >= K_CB) return;
    const float* row = emb + (size_t)wave * D_DIM;
    float s = 0.0f;
    #pragma unroll
    for (int i = lane * 4; i < D_DIM; i += 128) {
        float4 v = *(const float4*)(row + i);
        s += v.x * v.x + v.y * v.y + v.z * v.z + v.w * v.w;
    }
    #pragma unroll
    for (int m = 1; m < 32; m <<= 1) s += __shfl_xor(s, m, 32);
    if (lane == 0) ws[wave] = s;
}

// ---------------------------------------------------------------------------
// Kernel 2: main VQ. One block = 16 input rows vs all 8192 codewords.
// FP32 WMMA 16x16x4 accumulates dot(x,e); score = ||e||^2 - 2*dot.
// ---------------------------------------------------------------------------
__global__ __launch_bounds__(256) void vq_main(const float* __restrict__ x,
                                               const float* __restrict__ emb,
                                               const float* __restrict__ eNorm,
                                               float* __restrict__ outQ,
                                               float* __restrict__ outIdx,
                                               float* __restrict__ lossAcc) {
    __shared__ float lds_x[16 * LDSX_ST];
    __shared__ float redv[8][16];
    __shared__ int   redi[8][16];
    __shared__ int   finalIdx[16];
    __shared__ float lsum[8];

    const int tid  = threadIdx.x;
    const int lane = tid & 31;
    const int wave = tid >> 5;
    const int row0 = blockIdx.x * 16;

    // Stage 16x256 x-tile into LDS (coalesced float4 loads).
    for (int i = tid; i < 16 * (D_DIM / 4); i += 256) {
        const int r = i >> 6;               // 64 float4 per row
        const int c = (i & 63) * 4;
        float4 v = *(const float4*)(x + (size_t)(row0 + r) * D_DIM + c);
        *(float4*)&lds_x[r * LDSX_ST + c] = v;
    }
    __syncthreads();

    // WMMA lane mapping (16x16x4 f32): lanes 0-15 carry K={k,k+1}, lanes 16-31 K={k+2,k+3}.
    const int nIdx = lane & 15;             // A: row M   / B: column N
    const int kb   = (lane >> 4) * 2;       // K sub-offset per half-wave

    float minv[8];
    int   mini[8];
    #pragma unroll
    for (int r = 0; r < 8; ++r) { minv[r] = 3.4e38f; mini[r] = 0; }

    const float* xrow = &lds_x[nIdx * LDSX_ST + kb];

    // 512 column tiles of 16 codewords; wave w handles tiles w, w+8, ...
    for (int ct = wave; ct < K_CB / 16; ct += 8) {
        const int col0 = ct * 16;
        const float* erow = emb + (size_t)(col0 + nIdx) * D_DIM + kb;

        v8f c = {0.f, 0.f, 0.f, 0.f, 0.f, 0.f, 0.f, 0.f};
        #pragma unroll 8
        for (int k = 0; k < D_DIM; k += 4) {
            v2f a = *(const v2f*)(xrow + k);   // LDS, 8B aligned
            v2f b = *(const v2f*)(erow + k);   // L2-resident embedding
            c = __builtin_amdgcn_wmma_f32_16x16x4_f32(
                    false, a, false, b, (short)0, c, false, false);
        }

        const float en  = eNorm[col0 + nIdx];
        const int   col = col0 + nIdx;
        #pragma unroll
        for (int r = 0; r < 8; ++r) {
            const float s = en - 2.0f * c[r];  // + ||x||^2 is row-constant: irrelevant to argmin
            if (s < minv[r]) { minv[r] = s; mini[r] = col; }
        }
    }

    // Butterfly min-reduce within each 16-lane half (rows 0-7 low half, 8-15 high half).
    #pragma unroll
    for (int m = 1; m < 16; m <<= 1) {
        #pragma unroll
        for (int r = 0; r < 8; ++r) {
            const float ov = __shfl_xor(minv[r], m, 32);
            const int   oi = __shfl_xor(mini[r], m, 32);
            if (ov < minv[r] || (ov == minv[r] && oi < mini[r])) { minv[r] = ov; mini[r] = oi; }
        }
    }
    if (lane == 0 || lane == 16) {
        const int rbase = (lane >> 4) * 8;
        #pragma unroll
        for (int r = 0; r < 8; ++r) {
            redv[wave][rbase + r] = minv[r];
            redi[wave][rbase + r] = mini[r];
        }
    }
    __syncthreads();

    // Cross-wave reduce; write indices (argmin tie -> lowest index).
    if (tid < 16) {
        float bv = redv[0][tid];
        int   bi = redi[0][tid];
        #pragma unroll
        for (int w = 1; w < 8; ++w) {
            const float v = redv[w][tid];
            const int   i = redi[w][tid];
            if (v < bv || (v == bv && i < bi)) { bv = v; bi = i; }
        }
        finalIdx[tid]      = bi;
        outIdx[row0 + tid] = (float)bi;
    }
    __syncthreads();

    // Gather winning embedding rows -> outQ (straight-through forward == quantized),
    // accumulate sum((q - x)^2) for the loss.
    const int r    = tid >> 4;
    const int cseg = (tid & 15) * 16;
    const float* q  = emb + (size_t)finalIdx[r] * D_DIM + cseg;
    const float* xr = &lds_x[r * LDSX_ST + cseg];
    float* o = outQ + (size_t)(row0 + r) * D_DIM + cseg;

    float acc = 0.0f;
    #pragma unroll
    for (int j = 0; j < 16; j += 4) {
        float4 qv = *(const float4*)(q + j);
        *(float4*)(o + j) = qv;
        const float dx = qv.x - xr[j + 0];
        const float dy = qv.y - xr[j + 1];
        const float dz = qv.z - xr[j + 2];
        const float dw = qv.w - xr[j + 3];
        acc += dx * dx + dy * dy + dz * dz + dw * dw;
    }
    #pragma unroll
    for (int m = 1; m < 32; m <<= 1) acc += __shfl_xor(acc, m, 32);
    if (lane == 0) lsum[wave] = acc;
    __syncthreads();
    if (tid == 0) {
        float t = 0.0f;
        #pragma unroll
        for (int w = 0; w < 8; ++w) t += lsum[w];
        atomicAdd(lossAcc, t);
    }
}

// ---------------------------------------------------------------------------
// Kernel 3: loss = (q_latent + 0.25*e_latent) = 1.25 * mean((q-x)^2)
// ---------------------------------------------------------------------------
__global__ void vq_finalize(const float* __restrict__ lossAcc,
                            float* __restrict__ lossOut) {
    lossOut[0] = lossAcc[0] * (1.25f / (float)((size_t)N_TOT * D_DIM));
}

extern "C" void kernel_launch(void* const* d_in, const int* in_sizes, int n_in,
                              void* d_out, int out_size, void* d_ws, size_t ws_size,
                              hipStream_t stream) {
    (void)in_sizes; (void)n_in; (void)out_size; (void)ws_size;
    const float* x   = (const float*)d_in[0];   // [N, D] f32
    const float* emb = (const float*)d_in[1];   // [K, D] f32

    float* ws      = (float*)d_ws;              // [0..K) norms, [K] loss accum
    float* outQ    = (float*)d_out;             // [N*D] quantized_st
    float* lossOut = outQ + (size_t)N_TOT * D_DIM;  // [1] loss
    float* outIdx  = lossOut + 1;               // [N] indices (as float values)

    vq_norms   <<<K_CB / 8,  256, 0, stream>>>(emb, ws);
    vq_main    <<<N_TOT / 16, 256, 0, stream>>>(x, emb, ws, outQ, outIdx, ws + K_CB);
    vq_finalize<<<1, 1, 0, stream>>>(ws + K_CB, lossOut);
}